// SelfSimilarity_48215302865052
// MI455X (gfx1250) — compile-verified
//
#include <hip/hip_runtime.h>

typedef __attribute__((ext_vector_type(2))) float v2f;
typedef __attribute__((ext_vector_type(4))) float v4f;
typedef __attribute__((ext_vector_type(8))) float v8f;

#define NPTS 4096
#define DDIM 64
#define ROWS 16                         // rows per workgroup strip
#define NWAVES 8
#define TILES_PER_WAVE (NPTS / 16 / NWAVES)  // 32

__global__ __launch_bounds__(256, 1)
void selfsim_softmax_kernel(const float* __restrict__ x, float* __restrict__ out) {
    // 16 rows x 4096 scores, fp32: 256 KB of LDS (CDNA5 WGP has 320 KB)
    __shared__ float scores[ROWS * NPTS];
    __shared__ float rsum[ROWS];

    const int tid  = (int)threadIdx.x;
    const int lane = tid & 31;
    const int wave = tid >> 5;
    const int b    = (int)blockIdx.y;
    const int n0   = (int)blockIdx.x * ROWS;

    const float invT = 1.0f / 13.544f;
    const float* X = x + (size_t)b * NPTS * DDIM;

    // WMMA f32 16x16x4 fragment addressing (wave32):
    //   lane l holds row (l & 15); VGPR pair covers K = 4*kc + 2*(l>>4) + {0,1}
    const int mrow  = lane & 15;
    const int khalf = (lane >> 4) * 2;

    // ---- Preload the A fragments for this 16-row strip (16 x float2 = 32 VGPRs)
    v2f af[16];
    const float* arow = X + (size_t)(n0 + mrow) * DDIM + khalf;
#pragma unroll
    for (int kc = 0; kc < 16; ++kc)
        af[kc] = *(const v2f*)(arow + kc * 4);

    // ---- Phase 1: each wave computes 32 column tiles of the 16x4096 score strip
    for (int t = 0; t < TILES_PER_WAVE; ++t) {
        const int m0 = (wave * TILES_PER_WAVE + t) * 16;
        const float* brow = X + (size_t)(m0 + mrow) * DDIM + khalf;

        v8f acc = {};
        float sp = 0.0f;                 // partial sum-of-squares for this lane's column
#pragma unroll
        for (int kc = 0; kc < 16; ++kc) {
            v2f bf = *(const v2f*)(brow + kc * 4);
            sp += bf.x * bf.x + bf.y * bf.y;
            // D = A(16x4,f32) * B(4x16,f32) + C ; accumulate over K = 64
            acc = __builtin_amdgcn_wmma_f32_16x16x4_f32(
                false, af[kc], false, bf, (short)0, acc, false, false);
        }
        // full ||x_m||^2: this lane holds half the K range, partner lane^16 the rest
        const float sqm = sp + __shfl_xor(sp, 16, 32);

        // C/D layout: lane gives column N = lane&15; VGPR r gives row r (+8 for hi half)
        const int col   = m0 + mrow;
        const int rbase = (lane >> 4) << 3;
#pragma unroll
        for (int r = 0; r < 8; ++r) {
            // softmax logits: (2G - ||x_m||^2)/T  (row-constant ||x_n||^2 cancels)
            scores[(rbase + r) * NPTS + col] = (2.0f * acc[r] - sqm) * invT;
        }
    }
    __syncthreads();

    // ---- Phase 2: per-row max and sum of exp (16 threads per row, shuffle-reduce)
    {
        const int row = tid >> 4;
        const int i   = tid & 15;
        float vmax = -INFINITY;
        for (int m = i; m < NPTS; m += 16)
            vmax = fmaxf(vmax, scores[row * NPTS + m]);
#pragma unroll
        for (int off = 8; off > 0; off >>= 1)
            vmax = fmaxf(vmax, __shfl_xor(vmax, off, 32));

        float vsum = 0.0f;
        for (int m = i; m < NPTS; m += 16) {
            float e = __expf(scores[row * NPTS + m] - vmax);
            scores[row * NPTS + m] = e;      // overwrite with exp (unique owner)
            vsum += e;
        }
#pragma unroll
        for (int off = 8; off > 0; off >>= 1)
            vsum += __shfl_xor(vsum, off, 32);
        if (i == 0) rsum[row] = vsum;
    }
    __syncthreads();

    // ---- Phase 3: normalize and stream out with coalesced 16B stores
    const v4f* s4 = (const v4f*)scores;
    for (int r = 0; r < ROWS; ++r) {
        const float inv = 1.0f / rsum[r];
        float* orow = out + ((size_t)b * NPTS + (n0 + r)) * NPTS;
        for (int j = tid; j < NPTS / 4; j += 256) {
            v4f v = s4[r * (NPTS / 4) + j];
            v.x *= inv; v.y *= inv; v.z *= inv; v.w *= inv;
            *(v4f*)(orow + j * 4) = v;
        }
    }
}

extern "C" void kernel_launch(void* const* d_in, const int* in_sizes, int n_in,
                              void* d_out, int out_size, void* d_ws, size_t ws_size,
                              hipStream_t stream) {
    const float* x = (const float*)d_in[0];
    float* out = (float*)d_out;
    dim3 grid(NPTS / ROWS, 4);   // 256 row strips x 4 batches
    dim3 block(256);             // 8 waves (wave32)
    selfsim_softmax_kernel<<<grid, block, 0, stream>>>(x, out);
}